// ODEBlock_59133109731779
// MI455X (gfx1250) — compile-verified
//
#include <hip/hip_runtime.h>

// ---------------------------------------------------------------------------
// Coupled-STO LLG RK4 integrator for MI455X (gfx1250, wave32).
// One thread per (batch, oscillator): 8192*8 = 65536 threads.
// Lanes [8b, 8b+7] of each wave32 hold the 8 oscillators of one batch; the
// mean-field coupling sum is a 3-stage DPP butterfly (pure VALU, no LDS pipe).
// m0 tile (3 KB / block) staged global->LDS via the Tensor Data Mover.
// ---------------------------------------------------------------------------

typedef __attribute__((ext_vector_type(4))) unsigned int u32x4;
typedef __attribute__((ext_vector_type(8))) int          i32x8;
typedef __attribute__((ext_vector_type(4))) int          i32x4;

namespace {
constexpr float ALPHA   = 0.009f;
constexpr float H_ANIS  = 0.5f;
constexpr float CPL     = 0.1f / 7.0f;                 // COUPLING/(S-1)
constexpr float DT      = 0.01f;
constexpr float INV     = 1.0f / (1.0f + ALPHA * ALPHA);
// Fold the 1/(1+a^2) divisor of the derivative into the RK4 step constants.
constexpr float A_HALF  = 0.5f * DT * INV;             // g -> 0.5*DT*k
constexpr float A_FULL  = DT * INV;                    // g -> DT*k
constexpr float A_SIXTH = (DT / 6.0f) * INV;           // g -> (DT/6)*k

constexpr int TILE_ELEMS = 32 * 8 * 3;                 // floats per block tile (768)
constexpr int TILE_BYTES = TILE_ELEMS * 4;             // 3072 B
}

// v + cross-lane(v) with a DPP-sourced operand; stays entirely on the VALU.
template <int CTRL>
__device__ __forceinline__ float dpp_add(float v) {
    int s = __builtin_amdgcn_update_dpp(0, __float_as_int(v), CTRL,
                                        0xf, 0xf, true);
    return v + __int_as_float(s);
}

// Sum over each aligned 8-lane group (the 8 oscillators of one batch).
// Stage 1: quad_perm [1,0,3,2]  -> pair sums          (xor 1)
// Stage 2: quad_perm [2,3,0,1]  -> quad sums          (xor 2)
// Stage 3: row_half_mirror      -> add the other quad (lane ^ 7)
__device__ __forceinline__ float sum8(float v) {
    v = dpp_add<0x0B1>(v);
    v = dpp_add<0x04E>(v);
    v = dpp_add<0x141>(v);
    return v;
}

// Unscaled LLG derivative g; true k = INV * g (INV folded into step consts).
// h = h_app*ez + H_ANIS*m_x*ex + CPL*(sum_group - m)
// g = -(m x h) - ALPHA * m x (m x h) + j * m x (m x ex)
// with m x (m x ex) = ( -(y^2+z^2), x*y, x*z ).
__device__ __forceinline__ void llg_g(float x, float y, float z,
                                      float h_app, float j,
                                      float& gx, float& gy, float& gz)
{
    float sx = sum8(x);
    float sy = sum8(y);
    float sz = sum8(z);
    float hx = fmaf(H_ANIS, x, CPL * (sx - x));
    float hy = CPL * (sy - y);
    float hz = fmaf(CPL, sz - z, h_app);
    // a = m x h
    float ax = fmaf(y, hz, -z * hy);
    float ay = fmaf(z, hx, -x * hz);
    float az = fmaf(x, hy, -y * hx);
    // t = m x a
    float tx = fmaf(y, az, -z * ay);
    float ty = fmaf(z, ax, -x * az);
    float tz = fmaf(x, ay, -y * ax);
    gx = fmaf(-j, fmaf(y, y, z * z), fmaf(-ALPHA, tx, -ax));
    gy = fmaf( j, x * y,             fmaf(-ALPHA, ty, -ay));
    gz = fmaf( j, x * z,             fmaf(-ALPHA, tz, -az));
}

__global__ __launch_bounds__(256)
void sto_rk4_kernel(const float* __restrict__ m0,
                    const float* __restrict__ applied,
                    const float* __restrict__ current,
                    const int*   __restrict__ nsteps_p,
                    float*       __restrict__ out)
{
    __shared__ float smem[TILE_ELEMS];        // 32 batches x 8 STO x 3 comps
    const int tid = threadIdx.x;

    // ---- Stage this block's m0 tile into LDS with the Tensor Data Mover ----
    if (tid < 32) {   // wave 0 issues the DMA; EXEC is ignored by TDM anyway
        unsigned long long ga = (unsigned long long)(const void*)m0
                              + (unsigned long long)blockIdx.x * (unsigned long long)TILE_BYTES;
        u32x4 g0;
        g0[0] = 1u;                                        // count=1 valid descriptor
        g0[1] = 0u;                                        // lds_addr = 0 (start of smem)
        g0[2] = (unsigned)(ga & 0xFFFFFFFFu);              // global_addr[31:0]
        g0[3] = (unsigned)((ga >> 32) & 0x01FFFFFFu)       // global_addr[56:32]
              | (2u << 30);                                // type = 2 ("image")
        i32x8 d1;
        d1[0] = (2 << 16);                                 // data_size = 4 bytes
        d1[1] = (int)((TILE_ELEMS & 0xFFFF) << 16);        // tensor_dim0 lo16
        d1[2] = (1 << 16);                                 // dim0 hi16=0 | tensor_dim1 = 1
        d1[3] = (int)((unsigned)TILE_ELEMS << 16);         // dim1 hi16=0 | tile_dim0 = 768
        d1[4] = 0;                                         // tile_dim1/2 unused (1-D tile)
        d1[5] = TILE_ELEMS;                                // tensor_dim0_stride lo32
        d1[6] = 0;
        d1[7] = 0;
        i32x4 d2 = {0, 0, 0, 0};
        i32x4 d3 = {0, 0, 0, 0};
        i32x8 d4 = {0, 0, 0, 0, 0, 0, 0, 0};
        __builtin_amdgcn_tensor_load_to_lds(g0, d1, d2, d3, d4, 0);
        __builtin_amdgcn_s_wait_tensorcnt(0);
    }
    __syncthreads();

    // ---- Load my oscillator state from LDS and normalize onto the sphere ----
    float x = smem[tid * 3 + 0];
    float y = smem[tid * 3 + 1];
    float z = smem[tid * 3 + 2];
    float rn = 1.0f / sqrtf(fmaf(x, x, fmaf(y, y, z * z)));
    x *= rn; y *= rn; z *= rn;

    const int   gbatch = (blockIdx.x << 5) + (tid >> 3);   // global batch index
    const float h_app  = fmaf(0.5f, applied[gbatch], 1.5f);
    const float jcur   = fmaf(0.5f, current[gbatch], 0.25f);
    const int   nsteps = nsteps_p[0];

    // ---- 500 RK4 steps, fully in registers, DS-free inner loop ----
#pragma unroll 1
    for (int it = 0; it < nsteps; ++it) {
        float g1x, g1y, g1z, g2x, g2y, g2z, g3x, g3y, g3z, g4x, g4y, g4z;
        llg_g(x, y, z, h_app, jcur, g1x, g1y, g1z);
        llg_g(fmaf(A_HALF, g1x, x), fmaf(A_HALF, g1y, y), fmaf(A_HALF, g1z, z),
              h_app, jcur, g2x, g2y, g2z);
        llg_g(fmaf(A_HALF, g2x, x), fmaf(A_HALF, g2y, y), fmaf(A_HALF, g2z, z),
              h_app, jcur, g3x, g3y, g3z);
        llg_g(fmaf(A_FULL, g3x, x), fmaf(A_FULL, g3y, y), fmaf(A_FULL, g3z, z),
              h_app, jcur, g4x, g4y, g4z);
        x = fmaf(A_SIXTH, g1x + 2.0f * (g2x + g3x) + g4x, x);
        y = fmaf(A_SIXTH, g1y + 2.0f * (g2y + g3y) + g4y, y);
        z = fmaf(A_SIXTH, g1z + 2.0f * (g2z + g3z) + g4z, z);
    }

    const int base = (blockIdx.x * 256 + tid) * 3;
    out[base + 0] = x;
    out[base + 1] = y;
    out[base + 2] = z;
}

extern "C" void kernel_launch(void* const* d_in, const int* in_sizes, int n_in,
                              void* d_out, int out_size, void* d_ws, size_t ws_size,
                              hipStream_t stream) {
    (void)n_in; (void)out_size; (void)d_ws; (void)ws_size;
    const float* m0      = (const float*)d_in[0];   // [B,S,3] f32
    const float* applied = (const float*)d_in[1];   // [B,1,1] f32
    const float* current = (const float*)d_in[2];   // [B,1,1] f32
    const int*   nsteps  = (const int*)d_in[3];     // scalar (device)
    float*       out     = (float*)d_out;           // [B,S,3] f32

    const int total_threads = in_sizes[0] / 3;      // B*S = 65536
    dim3 block(256);
    dim3 grid(total_threads / 256);                 // 256 blocks of 32 batches
    sto_rk4_kernel<<<grid, block, 0, stream>>>(m0, applied, current, nsteps, out);
}